// NodeEdgeProjection_5025111736904
// MI455X (gfx1250) — compile-verified
//
#include <hip/hip_runtime.h>

// out[b, e, f] = in[b, e / 127, f]   (B=256, N=128, F=64, E = N*(N-1) = 16256)
// Pure broadcast: node row r (256 B) -> 127 consecutive output rows.
// Store-bandwidth bound: ~1.065 GB written => ~46 us floor @ 23.3 TB/s.

typedef float v4f __attribute__((ext_vector_type(4)));
typedef int   v4i __attribute__((ext_vector_type(4)));

#define BATCH   256
#define N_NODES 128
#define DEG     127            // output rows per node = N-1
#define FEAT    64             // floats per row
#define ROW_V4  (FEAT / 4)     // 16 float4 per 256-byte row

__global__ __launch_bounds__(256)
void node_edge_broadcast_kernel(const float* __restrict__ in,
                                float* __restrict__ out) {
    const int blk  = blockIdx.x;            // b * N_NODES + r
    const int t    = threadIdx.x;           // 0..255
    const int col  = t & (ROW_V4 - 1);      // float4 column within row: 0..15
    const int row0 = t >> 4;                // starting replica row: 0..15

    __shared__ v4f row[ROW_V4];             // one 256-byte node row

    const v4f* srcv = (const v4f*)in + (size_t)blk * ROW_V4;

#if __has_builtin(__builtin_amdgcn_global_load_async_to_lds_b128)
    // CDNA5 async data mover: stage the node row into LDS (ASYNCcnt-tracked).
    if (t < ROW_V4) {
        __builtin_amdgcn_global_load_async_to_lds_b128(
            (v4i*)(srcv + t),
            (v4i*)&row[t],
            /*offset=*/0, /*cpol=*/0);
    }
#if __has_builtin(__builtin_amdgcn_s_wait_asynccnt)
    __builtin_amdgcn_s_wait_asynccnt(0);
#else
    asm volatile("s_wait_asynccnt 0" ::: "memory");
#endif
#else
    // Fallback: plain 128-bit load through L2 (input is only 8 MB, L2-resident).
    if (t < ROW_V4) row[t] = srcv[t];
#endif
    __syncthreads();

    // Each thread keeps its float4 of the row in VGPRs and replicates it.
    const v4f v = row[col];

    // Output base for this (b, r): rows [blk*DEG, blk*DEG + DEG).
    v4f* dstv = (v4f*)out + (size_t)blk * DEG * ROW_V4 + col;

    // 256 threads cover 16 replica rows per sweep; 127 rows => 8 sweeps
    // (last sweep partial: row0 == 15 writes only 7 rows).
    // Non-temporal: output is write-once, keep it out of L2's way.
#pragma unroll
    for (int k = 0; k < 7; ++k) {
        __builtin_nontemporal_store(v, dstv + (size_t)(row0 + 16 * k) * ROW_V4);
    }
    if (row0 < 15) {
        __builtin_nontemporal_store(v, dstv + (size_t)(row0 + 112) * ROW_V4);
    }
}

extern "C" void kernel_launch(void* const* d_in, const int* in_sizes, int n_in,
                              void* d_out, int out_size, void* d_ws, size_t ws_size,
                              hipStream_t stream) {
    (void)in_sizes; (void)n_in; (void)out_size; (void)d_ws; (void)ws_size;
    const float* in  = (const float*)d_in[0];
    float*       out = (float*)d_out;

    // One block per (batch, node) pair: 256 * 128 = 32768 blocks, 256 threads
    // (8 wave32s) each. Each block writes 127 * 256 B = ~32 KB, coalesced
    // 512 B per wave-iteration.
    node_edge_broadcast_kernel<<<dim3(BATCH * N_NODES), dim3(256), 0, stream>>>(in, out);
}